// GreedyGroupedRouter_49417893708016
// MI455X (gfx1250) — compile-verified
//
#include <hip/hip_runtime.h>
#include <hip/hip_bf16.h>

typedef float v4f __attribute__((ext_vector_type(4)));

#define N_EXPERTS 256
#define N_GROUPS  8
#define GRID_BLOCKS 2048
#define BLOCK_THREADS 256
#define WAVES_PER_BLOCK 8   // wave32
#define NBUF 4              // async pipeline depth (tokens in flight per wave)

__global__ void GreedyGroupedRouter_zero_hist(float* __restrict__ hist) {
    hist[threadIdx.x] = 0.0f;
}

__global__ __launch_bounds__(BLOCK_THREADS)
void GreedyGroupedRouter_main(const float* __restrict__ logits,
                              float* __restrict__ routing_w,   // [seq,256]
                              float* __restrict__ topk_w,      // [seq,8]
                              float* __restrict__ topk_id_f,   // [seq,8] ids as float
                              float* __restrict__ hist_g,      // [256]
                              int seq, int total_waves)
{
    // per-wave ring: 8 waves * 4 buffers * 256 floats = 32 KB
    __shared__ float buf[WAVES_PER_BLOCK * NBUF * N_EXPERTS];
    __shared__ unsigned hist_s[N_EXPERTS];

    const int tid  = threadIdx.x;
    const int lane = tid & 31;
    const int wave = tid >> 5;
    const int gwave = blockIdx.x * WAVES_PER_BLOCK + wave;

    hist_s[tid] = 0u;           // blockDim == 256: one bin per thread
    __syncthreads();

    float* wbuf = &buf[wave * (NBUF * N_EXPERTS)];
    const unsigned lds_base = (unsigned)(unsigned long long)(&wbuf[lane * 8]);

    const int t0 = gwave;
    if (t0 < seq) {
        // last token this wave will process (clamp target for tail issues; stays L2-hot)
        const int lastT = t0 + ((seq - 1 - t0) / total_waves) * total_waves;

        // ---- prologue: fill the 4-deep ring (2 x b128 per lane per token = 1 KB/wave) ----
        #pragma unroll
        for (int k = 0; k < NBUF; ++k) {
            int ti = t0 + k * total_waves;
            if (ti > lastT) ti = lastT;
            unsigned ldsk = lds_base + (unsigned)k * (N_EXPERTS * 4u);
            unsigned long long g =
                (unsigned long long)(logits + (size_t)ti * N_EXPERTS + lane * 8);
            asm volatile("global_load_async_to_lds_b128 %0, %1, off\n\t"
                         "global_load_async_to_lds_b128 %0, %1, off offset:16"
                         :: "v"(ldsk), "v"(g) : "memory");
        }

        int i = 0;  // token ordinal -> ring slot i & 3
        for (int t = t0; t <= lastT; t += total_waves, ++i) {
            const int bufsel = i & (NBUF - 1);

            // oldest token ready when only the 3 newer tokens (6 loads) remain outstanding
            asm volatile("s_wait_asynccnt 0x6" ::: "memory");

            const float* cur = wbuf + bufsel * N_EXPERTS + lane * 8;
            v4f a = *(const v4f*)(cur);
            v4f b = *(const v4f*)(cur + 4);

            // refill this ring slot with token t + 4*stride (clamped); reads above are
            // anchored before this asm by the memory clobber, and the async write
            // cannot land before the ds_loads complete (HBM round-trip).
            {
                int ti = t + NBUF * total_waves;
                if (ti > lastT) ti = lastT;
                unsigned ldsk = lds_base + (unsigned)bufsel * (N_EXPERTS * 4u);
                unsigned long long g =
                    (unsigned long long)(logits + (size_t)ti * N_EXPERTS + lane * 8);
                asm volatile("global_load_async_to_lds_b128 %0, %1, off\n\t"
                             "global_load_async_to_lds_b128 %0, %1, off offset:16"
                             :: "v"(ldsk), "v"(g) : "memory");
            }

            float r[8] = {a.x, a.y, a.z, a.w, b.x, b.y, b.z, b.w};

            // ---- softmax max (wave32 xor reduction) ----
            float m = r[0];
            #pragma unroll
            for (int j = 1; j < 8; ++j) m = fmaxf(m, r[j]);
            #pragma unroll
            for (int s = 1; s < 32; s <<= 1) m = fmaxf(m, __shfl_xor(m, s, 32));

            float e[8], lsum = 0.0f;
            #pragma unroll
            for (int j = 0; j < 8; ++j) { e[j] = __expf(r[j] - m); lsum += e[j]; }
            float tot = lsum;
            #pragma unroll
            for (int s = 1; s < 32; s <<= 1) tot += __shfl_xor(tot, s, 32);
            const float inv = 1.0f / tot;

            float w[8];
            #pragma unroll
            for (int j = 0; j < 8; ++j) w[j] = e[j] * inv;

            // stream out routing weights (write-once 512 MB stream -> nontemporal)
            float* dst = routing_w + (size_t)t * N_EXPERTS + lane * 8;
            v4f wa = {w[0], w[1], w[2], w[3]};
            v4f wb = {w[4], w[5], w[6], w[7]};
            __builtin_nontemporal_store(wa, (v4f*)dst);
            __builtin_nontemporal_store(wb, (v4f*)(dst + 4));

            // ---- grouped top-1: lane's 8 experts all lie in group (lane>>2) ----
            float v = w[0];
            int   idx = lane * 8;
            #pragma unroll
            for (int j = 1; j < 8; ++j)
                if (w[j] > v) { v = w[j]; idx = lane * 8 + j; }   // strict >: first occurrence
            #pragma unroll
            for (int s = 1; s < 4; s <<= 1) {                     // reduce the 4 lanes of a group
                float ov = __shfl_xor(v, s, 32);
                int   oi = __shfl_xor(idx, s, 32);
                if (ov > v || (ov == v && oi < idx)) { v = ov; idx = oi; }
            }

            // every lane now holds its group's max -> wave-sum / 4 = sum of 8 selected
            float ssum = v;
            #pragma unroll
            for (int s = 1; s < 32; s <<= 1) ssum += __shfl_xor(ssum, s, 32);
            const float norm = 1.0f / (ssum * 0.25f + 1e-20f);

            if ((lane & 3) == 0) {
                int g = lane >> 2;
                size_t o = (size_t)t * N_GROUPS + g;
                topk_w[o]    = v * norm;
                topk_id_f[o] = (float)idx;
                atomicAdd(&hist_s[idx], 1u);
            }
        }
    }

    // drain outstanding async loads before the block-level histogram flush
    asm volatile("s_wait_asynccnt 0x0" ::: "memory");

    __syncthreads();
    unsigned c = hist_s[tid];
    if (c) atomicAdd(&hist_g[tid], (float)c);
}

extern "C" void kernel_launch(void* const* d_in, const int* in_sizes, int n_in,
                              void* d_out, int out_size, void* d_ws, size_t ws_size,
                              hipStream_t stream) {
    (void)n_in; (void)d_ws; (void)ws_size; (void)out_size;
    const float* logits = (const float*)d_in[0];
    const int seq = in_sizes[0] / N_EXPERTS;

    float* out       = (float*)d_out;
    float* routing_w = out;                                   // seq*256
    float* topk_w    = routing_w + (size_t)seq * N_EXPERTS;   // seq*8
    float* topk_id_f = topk_w    + (size_t)seq * N_GROUPS;    // seq*8 (ids as floats)
    float* hist_g    = topk_id_f + (size_t)seq * N_GROUPS;    // 256

    // zero the histogram every launch (atomics accumulate; replays must be deterministic)
    GreedyGroupedRouter_zero_hist<<<1, N_EXPERTS, 0, stream>>>(hist_g);

    const int total_waves = GRID_BLOCKS * WAVES_PER_BLOCK;
    GreedyGroupedRouter_main<<<GRID_BLOCKS, BLOCK_THREADS, 0, stream>>>(
        logits, routing_w, topk_w, topk_id_f, hist_g, seq, total_waves);
}